// Embedding_59124519797196
// MI455X (gfx1250) — compile-verified
//
#include <hip/hip_runtime.h>

// Embedding gather: out[row,:] = weight[idx[row],:], row in [0, B*S)
//   idxes : int32 [4, 2048]          (in_sizes[0] = 8192)
//   weight: f32   [50257, 1024]
//   out   : f32   [4, 2048, 1024]
//
// Pure HBM-bound copy (64 MB total traffic -> ~2.75 us @ 23.3 TB/s).
// WMMA would turn this into an 845-GFLOP one-hot matmul: architecturally wrong.
// Instead we exercise the CDNA5 async data-mover path:
//   global_load_async_to_lds_b128  (global -> LDS, tracked by ASYNCcnt)
//   s_wait_asynccnt 0
//   global_store_async_from_lds_b128 (LDS -> global)
// Each wave owns a 4 KB LDS slot == one embedding row; 32 lanes x 16 B = 512 B
// per async instruction, 8 instructions per row.

#define FEATURES        1024
#define WAVES_PER_BLOCK 8
#define THREADS         (WAVES_PER_BLOCK * 32)

// Address-space-qualified pointer types for the inline asm operands.
typedef __attribute__((address_space(3))) void       lds_t;     // 32-bit LDS addr
typedef __attribute__((address_space(1))) const void gconst_t;  // 64-bit global addr
typedef __attribute__((address_space(1))) void       g_t;

__global__ __launch_bounds__(THREADS)
void emb_gather_async(const int* __restrict__ idx,
                      const float* __restrict__ weight,
                      float* __restrict__ out,
                      int nrows)
{
    __shared__ __attribute__((aligned(16))) float lds_buf[WAVES_PER_BLOCK * FEATURES];

    const int lane = threadIdx.x & 31;
    const int wave = threadIdx.x >> 5;

    float* slot = &lds_buf[wave * FEATURES];   // private 4 KB slot per wave

    for (int row = blockIdx.x * WAVES_PER_BLOCK + wave; row < nrows;
         row += gridDim.x * WAVES_PER_BLOCK) {

        const int page = idx[row];                         // wave-uniform
        const float* src = weight + (size_t)page * FEATURES;
        float*       dst = out    + (size_t)row  * FEATURES;

        // ---- global -> LDS : 8 x async B128 (512 B per issue per wave) ----
#pragma unroll
        for (int s = 0; s < 8; ++s) {
            gconst_t* gp = (gconst_t*)(src + s * 128 + lane * 4);
            lds_t*    lp = (lds_t*)(slot + s * 128 + lane * 4);
            asm volatile("global_load_async_to_lds_b128 %0, %1, off"
                         :: "v"(lp), "v"(gp) : "memory");
        }
        // Loads complete (LDS written) when ASYNCcnt drains.
        asm volatile("s_wait_asynccnt 0" ::: "memory");

        // ---- LDS -> global : 8 x async B128 ----
#pragma unroll
        for (int s = 0; s < 8; ++s) {
            g_t*   gp = (g_t*)(dst + s * 128 + lane * 4);
            lds_t* lp = (lds_t*)(slot + s * 128 + lane * 4);
            asm volatile("global_store_async_from_lds_b128 %0, %1, off"
                         :: "v"(gp), "v"(lp) : "memory");
        }
        // Stores must have consumed the LDS slot before next iteration's
        // loads overwrite it. (s_endpgm also implies wait-idle at kernel end.)
        asm volatile("s_wait_asynccnt 0" ::: "memory");
    }
}

extern "C" void kernel_launch(void* const* d_in, const int* in_sizes, int n_in,
                              void* d_out, int out_size, void* d_ws, size_t ws_size,
                              hipStream_t stream) {
    const int*   idx    = (const int*)d_in[0];    // int32 [4,2048]
    const float* weight = (const float*)d_in[1];  // f32 [50257,1024]
    float*       out    = (float*)d_out;          // f32 [4,2048,1024]

    const int nrows  = in_sizes[0];               // 8192 rows
    const int blocks = (nrows + WAVES_PER_BLOCK - 1) / WAVES_PER_BLOCK;  // 1024

    emb_gather_async<<<blocks, THREADS, 0, stream>>>(idx, weight, out, nrows);
}